// CRPS_5634997092912
// MI455X (gfx1250) — compile-verified
//
#include <hip/hip_runtime.h>
#include <hip/hip_bf16.h>
#include <stdint.h>

#define NBINS      1000
#define NBINS_PAD  1024
#define NCOLS      64

typedef float v2f __attribute__((ext_vector_type(2)));
typedef float v8f __attribute__((ext_vector_type(8)));

#if defined(__has_builtin)
#if __has_builtin(__builtin_amdgcn_wmma_f32_16x16x4_f32)
#define HAVE_WMMA_F32X4 1
#endif
#endif

// Order-preserving float <-> uint mapping (for atomicMin/Max on encoded bits).
__device__ __forceinline__ unsigned enc_f32(float f) {
  unsigned u = __float_as_uint(f);
  return (u & 0x80000000u) ? ~u : (u | 0x80000000u);
}
__device__ __forceinline__ float dec_f32(unsigned u) {
  u = (u & 0x80000000u) ? (u & 0x7FFFFFFFu) : ~u;
  return __uint_as_float(u);
}

// ws layout (uint32): [0]=min_enc, [1]=max_enc, [2..1025]=hist_p, [1026..2049]=hist_t
__global__ void crps_init(unsigned* __restrict__ ws) {
  if (blockIdx.x == 0 && threadIdx.x == 0) { ws[0] = 0xFFFFFFFFu; ws[1] = 0u; }
  int stride = gridDim.x * blockDim.x;
  for (int j = blockIdx.x * blockDim.x + threadIdx.x; j < 2 * NBINS_PAD; j += stride)
    ws[2 + j] = 0u;
}

__global__ void crps_minmax(const float* __restrict__ p, const float* __restrict__ t,
                            long long n, unsigned* __restrict__ ws) {
  long long gid    = (long long)blockIdx.x * blockDim.x + threadIdx.x;
  long long stride = (long long)gridDim.x * blockDim.x;
  long long n4 = n >> 2;
  float lmin =  3.402823466e38f;
  float lmax = -3.402823466e38f;
  const float4* p4 = (const float4*)p;
  const float4* t4 = (const float4*)t;
  for (long long i = gid; i < n4; i += stride) {          // global_load_b128
    float4 a = p4[i], b = t4[i];
    lmin = fminf(lmin, fminf(fminf(a.x, a.y), fminf(a.z, a.w)));
    lmax = fmaxf(lmax, fmaxf(fmaxf(a.x, a.y), fmaxf(a.z, a.w)));
    lmin = fminf(lmin, fminf(fminf(b.x, b.y), fminf(b.z, b.w)));
    lmax = fmaxf(lmax, fmaxf(fmaxf(b.x, b.y), fmaxf(b.z, b.w)));
  }
  for (long long i = (n4 << 2) + gid; i < n; i += stride) { // tail
    lmin = fminf(lmin, fminf(p[i], t[i]));
    lmax = fmaxf(lmax, fmaxf(p[i], t[i]));
  }
  for (int off = 16; off > 0; off >>= 1) {                 // wave32 reduce
    lmin = fminf(lmin, __shfl_xor(lmin, off, 32));
    lmax = fmaxf(lmax, __shfl_xor(lmax, off, 32));
  }
  if ((threadIdx.x & 31) == 0) {
    atomicMin(&ws[0], enc_f32(lmin));
    atomicMax(&ws[1], enc_f32(lmax));
  }
}

__device__ __forceinline__ int bin_of(float v, float lo, float inv_dx) {
  int k = (int)ceilf((v - lo) * inv_dx);
  k = k < 0 ? 0 : k;
  return k > (NBINS - 1) ? (NBINS - 1) : k;
}

__global__ void crps_hist(const float* __restrict__ p, const float* __restrict__ t,
                          long long n, unsigned* __restrict__ ws) {
  __shared__ unsigned hp[NBINS_PAD];
  __shared__ unsigned ht[NBINS_PAD];
  for (int j = threadIdx.x; j < NBINS_PAD; j += blockDim.x) { hp[j] = 0u; ht[j] = 0u; }
  __syncthreads();

  float lo = dec_f32(ws[0]);
  float hi = dec_f32(ws[1]);
  float dx = (hi - lo) * (1.0f / (float)(NBINS - 1));
  float inv_dx = (dx > 0.0f) ? (1.0f / dx) : 0.0f;

  long long gid    = (long long)blockIdx.x * blockDim.x + threadIdx.x;
  long long stride = (long long)gridDim.x * blockDim.x;
  long long n4 = n >> 2;
  const float4* p4 = (const float4*)p;
  const float4* t4 = (const float4*)t;
  for (long long i = gid; i < n4; i += stride) {
    float4 a = p4[i], b = t4[i];
    atomicAdd(&hp[bin_of(a.x, lo, inv_dx)], 1u);           // ds_add_u32
    atomicAdd(&hp[bin_of(a.y, lo, inv_dx)], 1u);
    atomicAdd(&hp[bin_of(a.z, lo, inv_dx)], 1u);
    atomicAdd(&hp[bin_of(a.w, lo, inv_dx)], 1u);
    atomicAdd(&ht[bin_of(b.x, lo, inv_dx)], 1u);
    atomicAdd(&ht[bin_of(b.y, lo, inv_dx)], 1u);
    atomicAdd(&ht[bin_of(b.z, lo, inv_dx)], 1u);
    atomicAdd(&ht[bin_of(b.w, lo, inv_dx)], 1u);
  }
  for (long long i = (n4 << 2) + gid; i < n; i += stride) {
    atomicAdd(&hp[bin_of(p[i], lo, inv_dx)], 1u);
    atomicAdd(&ht[bin_of(t[i], lo, inv_dx)], 1u);
  }
  __syncthreads();
  unsigned* gp = ws + 2;
  unsigned* gt = ws + 2 + NBINS_PAD;
  for (int j = threadIdx.x; j < NBINS_PAD; j += blockDim.x) {
    unsigned cp = hp[j]; if (cp) atomicAdd(&gp[j], cp);
    unsigned ct = ht[j]; if (ct) atomicAdd(&gt[j], ct);
  }
}

// Inclusive prefix sum of 16-entry columns via D = L(16x16 lower-tri ones) x B,
// chained as 4x V_WMMA_F32_16X16X4_F32 (K=4 each). Exact: counts < 2^24.
// Tile layout: tile covers 256 bins; element (m,n) = hist[256*tile + 16*n + m].
__device__ __forceinline__ void prefix_tiles(const unsigned* __restrict__ hist,
                                             float* __restrict__ pre, int lane) {
  int nm    = lane & 15;   // row of A / column of B / column of D
  int khalf = lane >> 4;   // lane-half selects K pair (A,B) and M half (D)
#ifdef HAVE_WMMA_F32X4
  for (int tile = 0; tile < 4; ++tile) {
    int base = tile << 8;
    v8f acc = {0.f, 0.f, 0.f, 0.f, 0.f, 0.f, 0.f, 0.f};
#pragma unroll
    for (int j = 0; j < 4; ++j) {
      int ks = (j << 2) + (khalf << 1);        // k index = 4j + 2*lanehalf + reg
      v2f a, b;
      a.x = (ks     <= nm) ? 1.0f : 0.0f;      // L[m][k] = (k <= m)
      a.y = (ks + 1 <= nm) ? 1.0f : 0.0f;
      b.x = (float)hist[base + (nm << 4) + ks];      // B[k][n] = hist[base+16n+k]
      b.y = (float)hist[base + (nm << 4) + ks + 1];
      acc = __builtin_amdgcn_wmma_f32_16x16x4_f32(false, a, false, b,
                                                  (short)0, acc, false, false);
    }
#pragma unroll
    for (int r = 0; r < 8; ++r)                // D: m = r + 8*lanehalf, n = lane&15
      pre[base + (nm << 4) + (khalf << 3) + r] = acc[r];
  }
#else
  // Fallback (no f32x4 WMMA builtin): scalar per-column scans.
  for (int c = lane; c < NCOLS; c += 32) {
    float run = 0.0f;
    for (int m = 0; m < 16; ++m) {
      run += (float)hist[(c << 4) + m];
      pre[(c << 4) + m] = run;
    }
  }
#endif
}

__global__ void crps_final(const unsigned* __restrict__ ws, float* __restrict__ out,
                           long long n) {
  __shared__ float pre_p[NBINS_PAD];
  __shared__ float pre_t[NBINS_PAD];
  __shared__ float offs_p[NCOLS];
  __shared__ float offs_t[NCOLS];
  int lane = threadIdx.x;                       // single wave32, EXEC all ones
  float lo = dec_f32(ws[0]);
  float hi = dec_f32(ws[1]);
  float dx = (hi - lo) * (1.0f / (float)(NBINS - 1));

  prefix_tiles(ws + 2, pre_p, lane);
  prefix_tiles(ws + 2 + NBINS_PAD, pre_t, lane);
  __syncthreads();

  if (lane == 0) {                              // 64-column carry scan (tiny)
    float rp = 0.f, rt = 0.f;
    for (int c = 0; c < NCOLS; ++c) {
      offs_p[c] = rp; rp += pre_p[(c << 4) + 15];
      offs_t[c] = rt; rt += pre_t[(c << 4) + 15];
    }
  }
  __syncthreads();

  float invN = 1.0f / (float)n;
  float sum = 0.f;
  for (int k = lane; k < NBINS - 1; k += 32) {  // trapezoid over 999 intervals
    float cp0 = (pre_p[k]     + offs_p[k >> 4])       * invN;
    float ct0 = (pre_t[k]     + offs_t[k >> 4])       * invN;
    float cp1 = (pre_p[k + 1] + offs_p[(k + 1) >> 4]) * invN;
    float ct1 = (pre_t[k + 1] + offs_t[(k + 1) >> 4]) * invN;
    float d0 = fabsf(cp0 - ct0);
    float d1 = fabsf(cp1 - ct1);
    sum += 0.5f * (d0 * d0 + d1 * d1) * dx;
  }
  for (int off = 16; off > 0; off >>= 1) sum += __shfl_xor(sum, off, 32);
  if (lane == 0) out[0] = sum;
}

extern "C" void kernel_launch(void* const* d_in, const int* in_sizes, int n_in,
                              void* d_out, int out_size, void* d_ws, size_t ws_size,
                              hipStream_t stream) {
  (void)n_in; (void)out_size; (void)ws_size;
  const float* p = (const float*)d_in[0];
  const float* t = (const float*)d_in[1];
  long long n = (long long)in_sizes[0];
  unsigned* ws = (unsigned*)d_ws;
  float* out = (float*)d_out;

  crps_init<<<8, 256, 0, stream>>>(ws);
  crps_minmax<<<2048, 256, 0, stream>>>(p, t, n, ws);   // HBM-bound streaming pass
  crps_hist<<<1024, 256, 0, stream>>>(p, t, n, ws);     // likely L2-resident pass
  crps_final<<<1, 32, 0, stream>>>(ws, out, n);         // WMMA prefix + trapz
}